// GAT_77077483094553
// MI455X (gfx1250) — compile-verified
//
#include <hip/hip_runtime.h>
#include <math.h>

#define NN 100000
#define HEADS 4
#define NEG_SLOPE 0.2f

typedef float v2f __attribute__((ext_vector_type(2)));
typedef float v8f __attribute__((ext_vector_type(8)));

__device__ __forceinline__ float leakyf(float v) { return v > 0.f ? v : NEG_SLOPE * v; }

// float atomic max via signed/unsigned integer ordering trick (works for mixed signs,
// init value -inf = 0xff800000)
__device__ __forceinline__ void atomicMaxF(float* addr, float v) {
  if (v >= 0.f) atomicMax((int*)addr, __float_as_int(v));
  else          atomicMin((unsigned int*)addr, __float_as_uint(v));
}

// ---------------------------------------------------------------------------
// GEMM1: h1[N,16] = x[N,512] @ W1[512,16] using V_WMMA_F32_16X16X4_F32.
// One wave handles one 16-row M tile (N=16 fits in a single WMMA tile).
// A-frag (16x4): lane L<16 -> rows, VGPR0/1 = K=0/1; lanes 16-31 carry K=2/3.
// B-frag (4x16): mirror layout, lane = column, VGPRs stripe K.
// D (16x16 f32): VGPR v -> row M = v + (lane<16?0:8), col = lane&15.
// ---------------------------------------------------------------------------
__global__ void k_gemm1_wmma(const float* __restrict__ x,
                             const float* __restrict__ W,
                             float* __restrict__ h1, int nTiles) {
  int wave = (int)((blockIdx.x * blockDim.x + threadIdx.x) >> 5);
  if (wave >= nTiles) return;               // wave-uniform: EXEC stays all-ones
  int lane = threadIdx.x & 31;
  int row  = lane & 15;
  int koff = (lane >> 4) << 1;              // 0 or 2

  const float* xrow = x + (size_t)(wave * 16 + row) * 512;
  v8f acc = {};
  for (int k = 0; k < 512; k += 4) {
    v2f a, b;
    a.x = xrow[k + koff];
    a.y = xrow[k + koff + 1];
    b.x = W[(size_t)(k + koff) * 16 + row];
    b.y = W[(size_t)(k + koff + 1) * 16 + row];
    acc = __builtin_amdgcn_wmma_f32_16x16x4_f32(false, a, false, b,
                                                (short)0, acc, false, false);
  }
  int mbase = wave * 16 + ((lane >> 4) << 3);
  int col   = lane & 15;
#pragma unroll
  for (int v = 0; v < 8; ++v)
    h1[(size_t)(mbase + v) * 16 + col] = acc[v];
}

// ---------------------------------------------------------------------------
// Layer-1 prep: attention logits per node + init accumulators (every call,
// since the harness does not re-initialize d_ws between replays).
// ---------------------------------------------------------------------------
__global__ void k_prep1(const float* __restrict__ h1,
                        const float* __restrict__ a_src,   // [4,4]
                        const float* __restrict__ a_dst,   // [4,4]
                        float* __restrict__ als, float* __restrict__ ald,
                        float* __restrict__ m, float* __restrict__ s,
                        float* __restrict__ agg, int n_nodes) {
  int n = blockIdx.x * blockDim.x + threadIdx.x;
  if (n >= n_nodes) return;
#pragma unroll
  for (int h = 0; h < HEADS; ++h) {
    float vs = 0.f, vd = 0.f;
#pragma unroll
    for (int c = 0; c < 4; ++c) {
      float hv = h1[n * 16 + h * 4 + c];
      vs += hv * a_src[h * 4 + c];
      vd += hv * a_dst[h * 4 + c];
    }
    als[n * 4 + h] = vs;
    ald[n * 4 + h] = vd;
    m[n * 4 + h]   = -INFINITY;
    s[n * 4 + h]   = 0.f;
  }
#pragma unroll
  for (int j = 0; j < 16; ++j) agg[n * 16 + j] = 0.f;
}

// ---------------------------------------------------------------------------
// Edge pass A: segment max of leaky_relu(al_src[src] + al_dst[dst]) per head.
// Edge ids >= E are the implicit self-loops (src = dst = e - E).
// ---------------------------------------------------------------------------
__global__ void k_edge_max(const int* __restrict__ srcv, const int* __restrict__ dstv,
                           int E, int n_nodes,
                           const float* __restrict__ als, const float* __restrict__ ald,
                           float* __restrict__ m) {
  int e = blockIdx.x * blockDim.x + threadIdx.x;
  int ET = E + n_nodes;
  if (e >= ET) return;
  int sn = (e < E) ? srcv[e] : (e - E);
  int dn = (e < E) ? dstv[e] : (e - E);
#pragma unroll
  for (int h = 0; h < HEADS; ++h) {
    float v = leakyf(als[sn * 4 + h] + ald[dn * 4 + h]);
    atomicMaxF(&m[dn * 4 + h], v);
  }
}

// ---------------------------------------------------------------------------
// Edge pass B: w = exp(e - m[dst]); accumulate softmax denominator and the
// (unnormalized) weighted feature aggregate.  alpha = w/s distributes over
// the sum, so normalization happens once per node afterwards.
// ---------------------------------------------------------------------------
template <int C>
__global__ void k_edge_acc(const int* __restrict__ srcv, const int* __restrict__ dstv,
                           int E, int n_nodes,
                           const float* __restrict__ als, const float* __restrict__ ald,
                           const float* __restrict__ m,
                           const float* __restrict__ hfeat,  // [N, HEADS*C]
                           float* __restrict__ ssum, float* __restrict__ agg) {
  int e = blockIdx.x * blockDim.x + threadIdx.x;
  int ET = E + n_nodes;
  if (e >= ET) return;
  int sn = (e < E) ? srcv[e] : (e - E);
  int dn = (e < E) ? dstv[e] : (e - E);
#pragma unroll
  for (int h = 0; h < HEADS; ++h) {
    float v = leakyf(als[sn * 4 + h] + ald[dn * 4 + h]);
    float w = expf(v - m[dn * 4 + h]);
    atomicAdd(&ssum[dn * 4 + h], w);
#pragma unroll
    for (int c = 0; c < C; ++c)
      atomicAdd(&agg[(dn * HEADS + h) * C + c],
                hfeat[(sn * HEADS + h) * C + c] * w);
  }
}

// ---------------------------------------------------------------------------
// Layer-1 finalize: normalize, +bias, ELU (in place in agg -> act1).
// ---------------------------------------------------------------------------
__global__ void k_final1(float* __restrict__ agg, const float* __restrict__ s,
                         const float* __restrict__ b1, int n_nodes) {
  int n = blockIdx.x * blockDim.x + threadIdx.x;
  if (n >= n_nodes) return;
#pragma unroll
  for (int j = 0; j < 16; ++j) {
    int h = j >> 2;
    float v = agg[n * 16 + j] / (s[n * 4 + h] + 1e-16f) + b1[j];
    agg[n * 16 + j] = v > 0.f ? v : (expf(v) - 1.f);   // ELU
  }
}

// ---------------------------------------------------------------------------
// Layer-2 GEMM (tiny: [N,16]x[16,4]) fused with layer-2 prep + init.
// ---------------------------------------------------------------------------
__global__ void k_layer2(const float* __restrict__ act1,  // [N,16]
                         const float* __restrict__ W2,    // [16,4]
                         const float* __restrict__ a_src2, const float* __restrict__ a_dst2,
                         float* __restrict__ h2, float* __restrict__ als,
                         float* __restrict__ ald, float* __restrict__ m,
                         float* __restrict__ s, float* __restrict__ agg, int n_nodes) {
  int n = blockIdx.x * blockDim.x + threadIdx.x;
  if (n >= n_nodes) return;
  float a[16];
#pragma unroll
  for (int j = 0; j < 16; ++j) a[j] = act1[n * 16 + j];
#pragma unroll
  for (int h = 0; h < HEADS; ++h) {
    float v = 0.f;
#pragma unroll
    for (int j = 0; j < 16; ++j) v += a[j] * W2[j * 4 + h];
    h2[n * 4 + h]  = v;
    als[n * 4 + h] = v * a_src2[h];
    ald[n * 4 + h] = v * a_dst2[h];
    m[n * 4 + h]   = -INFINITY;
    s[n * 4 + h]   = 0.f;
    agg[n * 4 + h] = 0.f;
  }
}

// ---------------------------------------------------------------------------
// Layer-2 finalize fused with final FC: y = (agg/s + b2) @ Wfc + bfc.
// ---------------------------------------------------------------------------
__global__ void k_final2(const float* __restrict__ agg, const float* __restrict__ s,
                         const float* __restrict__ b2, const float* __restrict__ Wfc,
                         const float* __restrict__ bfc, float* __restrict__ out,
                         int n_nodes) {
  int n = blockIdx.x * blockDim.x + threadIdx.x;
  if (n >= n_nodes) return;
  float y = bfc[0];
#pragma unroll
  for (int h = 0; h < HEADS; ++h) {
    float o = agg[n * 4 + h] / (s[n * 4 + h] + 1e-16f) + b2[h];
    y += o * Wfc[h];
  }
  out[n] = y;
}

extern "C" void kernel_launch(void* const* d_in, const int* in_sizes, int n_in,
                              void* d_out, int out_size, void* d_ws, size_t ws_size,
                              hipStream_t stream) {
  const float* x      = (const float*)d_in[0];
  const int*   ei     = (const int*)d_in[1];   // [2, E] flat: row0 = src, row1 = dst
  const float* W1     = (const float*)d_in[2];
  const float* a_src1 = (const float*)d_in[3];
  const float* a_dst1 = (const float*)d_in[4];
  const float* b1     = (const float*)d_in[5];
  const float* W2     = (const float*)d_in[6];
  const float* a_src2 = (const float*)d_in[7];
  const float* a_dst2 = (const float*)d_in[8];
  const float* b2     = (const float*)d_in[9];
  const float* Wfc    = (const float*)d_in[10];
  const float* bfc    = (const float*)d_in[11];
  float*       out    = (float*)d_out;

  const int N = NN;
  const int E = in_sizes[1] / 2;
  const int ET = E + N;
  const int* srcv = ei;
  const int* dstv = ei + E;

  // workspace layout (floats); total = N*72 floats = 28.8 MB
  float* ws   = (float*)d_ws;
  float* h1   = ws;              // N*16
  float* als1 = h1 + (size_t)N * 16;   // N*4
  float* ald1 = als1 + (size_t)N * 4;  // N*4
  float* m1   = ald1 + (size_t)N * 4;  // N*4
  float* s1   = m1 + (size_t)N * 4;    // N*4
  float* agg1 = s1 + (size_t)N * 4;    // N*16 (becomes act1 in place)
  float* h2   = agg1 + (size_t)N * 16; // N*4
  float* als2 = h2 + (size_t)N * 4;    // N*4
  float* ald2 = als2 + (size_t)N * 4;  // N*4
  float* m2   = ald2 + (size_t)N * 4;  // N*4
  float* s2   = m2 + (size_t)N * 4;    // N*4
  float* agg2 = s2 + (size_t)N * 4;    // N*4

  const int nTiles = N / 16;                  // 6250 exactly
  dim3 blk(256);
  dim3 grdGemm((nTiles + 7) / 8);             // 8 waves / block
  dim3 grdNode((N + 255) / 256);
  dim3 grdEdge((ET + 255) / 256);

  // ---- Layer 1 ----
  k_gemm1_wmma<<<grdGemm, blk, 0, stream>>>(x, W1, h1, nTiles);
  k_prep1<<<grdNode, blk, 0, stream>>>(h1, a_src1, a_dst1, als1, ald1, m1, s1, agg1, N);
  k_edge_max<<<grdEdge, blk, 0, stream>>>(srcv, dstv, E, N, als1, ald1, m1);
  k_edge_acc<4><<<grdEdge, blk, 0, stream>>>(srcv, dstv, E, N, als1, ald1, m1, h1, s1, agg1);
  k_final1<<<grdNode, blk, 0, stream>>>(agg1, s1, b1, N);

  // ---- Layer 2 ----
  k_layer2<<<grdNode, blk, 0, stream>>>(agg1, W2, a_src2, a_dst2, h2, als2, ald2, m2, s2, agg2, N);
  k_edge_max<<<grdEdge, blk, 0, stream>>>(srcv, dstv, E, N, als2, ald2, m2);
  k_edge_acc<1><<<grdEdge, blk, 0, stream>>>(srcv, dstv, E, N, als2, ald2, m2, h2, s2, agg2);
  k_final2<<<grdNode, blk, 0, stream>>>(agg2, s2, b2, Wfc, bfc, out, N);
}